// GCN_66005057405150
// MI455X (gfx1250) — compile-verified
//
#include <hip/hip_runtime.h>
#include <hip/hip_bf16.h>

#define NN 100000
#define EE 1250000
#define FH 64
#define BB 256
#define CC 2
#define WLDS_STRIDE 65   // 64 + 1 pad: rows skew by one bank (half-wave banks still overlap)

typedef float v2f __attribute__((ext_vector_type(2)));
typedef float v8f __attribute__((ext_vector_type(8)));

// ---------------- degree / normalization ----------------

__global__ void gcn_zero_f32(float* __restrict__ p, int n) {
    int i = blockIdx.x * blockDim.x + threadIdx.x;
    if (i < n) p[i] = 0.0f;
}

__global__ void gcn_deg_accum(const int* __restrict__ dst, float* __restrict__ deg, int e) {
    int i = blockIdx.x * blockDim.x + threadIdx.x;
    if (i < e) atomicAdd(&deg[dst[i]], 1.0f);
}

__global__ void gcn_deg_finalize(float* __restrict__ deg, int n) {
    int i = blockIdx.x * blockDim.x + threadIdx.x;
    if (i < n) deg[i] = rsqrtf(deg[i] + 1.0f);   // becomes dinv_sqrt in place
}

// norm[e] = dinv[src]*dinv[dst], computed once and reused by all 3 layers
__global__ void gcn_edge_norm(const int* __restrict__ src, const int* __restrict__ dst,
                              const float* __restrict__ dinv, float* __restrict__ norm, int e) {
    int i = blockIdx.x * blockDim.x + threadIdx.x;
    if (i < e) norm[i] = dinv[src[i]] * dinv[dst[i]];
}

// ---------------- fp32 WMMA GEMM: out[nrows x 64] = A[nrows x 64] @ W[64 x 64] ----------------
// W (16KB) staged in LDS once per block (8 waves share it). One wave32 computes a
// 16x16 output tile, K marched in steps of 4 via v_wmma_f32_16x16x4_f32.
// nrows is a multiple of 16 (100000 = 6250*16).

__global__ void gcn_gemm64_wmma(const float* __restrict__ A,
                                const float* __restrict__ W,
                                float* __restrict__ out, int nrows) {
    __shared__ float Ws[FH * WLDS_STRIDE];
    for (int i = threadIdx.x; i < FH * FH; i += blockDim.x) {
        int r = i >> 6, c = i & 63;
        Ws[r * WLDS_STRIDE + c] = W[i];
    }
    __syncthreads();

    int gwave = blockIdx.x * (blockDim.x >> 5) + (threadIdx.x >> 5);
    int lane  = threadIdx.x & 31;
    int row0  = gwave * 16;
    if (row0 >= nrows) return;            // whole-wave uniform exit, EXEC stays full

    int half = lane >> 4;                 // 0: K pair {k0,k0+1}, 1: {k0+2,k0+3}
    int col  = lane & 15;                 // M for A-load, N for B-load / store

    v8f c = {0.f,0.f,0.f,0.f,0.f,0.f,0.f,0.f};
    const float* arow = A + (size_t)(row0 + col) * FH;

#pragma unroll
    for (int k0 = 0; k0 < FH; k0 += 4) {
        int ka = k0 + 2 * half;
        v2f a, b;
        a.x = arow[ka];
        a.y = arow[ka + 1];
        b.x = Ws[ka * WLDS_STRIDE + col];
        b.y = Ws[(ka + 1) * WLDS_STRIDE + col];
        c = __builtin_amdgcn_wmma_f32_16x16x4_f32(false, a, false, b,
                                                  (short)0, c, false, false);
    }

#pragma unroll
    for (int r = 0; r < 8; ++r) {
        int m = r + 8 * half;             // C/D layout: VGPR r -> M = r (+8 for hi lanes)
        out[(size_t)(row0 + m) * FH + col] = c[r];
    }
}

// ---------------- self-loop + bias init: agg = dinv^2 * h + b (float4, 16 thr/node) ---------

__global__ void gcn_agg_init(const float4* __restrict__ h4, const float* __restrict__ dinv,
                             const float4* __restrict__ bias4, float4* __restrict__ agg4, int n) {
    int t = blockIdx.x * blockDim.x + threadIdx.x;
    int node = t >> 4, f4 = t & 15;
    if (node >= n) return;
    float dv = dinv[node];
    float s = dv * dv;
    float4 v = h4[(size_t)node * 16 + f4];
    float4 b = bias4[f4];
    float4 o = { s * v.x + b.x, s * v.y + b.y, s * v.z + b.z, s * v.w + b.w };
    agg4[(size_t)node * 16 + f4] = o;
}

// ---------------- edge scatter: agg[dst] += norm[e] * h[src] ----------------
// 16 threads/edge, float4 gather (global_load_b128), 4 fp32 atomics per thread.
// Node matrix is 25.6MB << 192MB L2, so gathers and atomics are L2-resident.

__global__ void gcn_scatter(const float4* __restrict__ h4, const int* __restrict__ src,
                            const int* __restrict__ dst, const float* __restrict__ norm,
                            float* __restrict__ agg, int e) {
    long long t = (long long)blockIdx.x * blockDim.x + threadIdx.x;
    int edge = (int)(t >> 4), f4 = (int)(t & 15);
    if (edge >= e) return;
    int s = src[edge], d = dst[edge];
    float nm = norm[edge];
    float4 v = h4[(size_t)s * 16 + f4];
    float* ap = agg + (size_t)d * FH + f4 * 4;
    atomicAdd(ap + 0, nm * v.x);
    atomicAdd(ap + 1, nm * v.y);
    atomicAdd(ap + 2, nm * v.z);
    atomicAdd(ap + 3, nm * v.w);
}

__global__ void gcn_relu4(float4* __restrict__ p, int n4) {
    int i = blockIdx.x * blockDim.x + threadIdx.x;
    if (i >= n4) return;
    float4 v = p[i];
    v.x = fmaxf(v.x, 0.f); v.y = fmaxf(v.y, 0.f);
    v.z = fmaxf(v.z, 0.f); v.w = fmaxf(v.w, 0.f);
    p[i] = v;
}

// ---------------- pooling ----------------

__global__ void gcn_pool_accum(const float4* __restrict__ h4, const int* __restrict__ batch,
                               float* __restrict__ psum, float* __restrict__ pcnt, int n) {
    int t = blockIdx.x * blockDim.x + threadIdx.x;
    int node = t >> 4, f4 = t & 15;
    if (node >= n) return;
    int b = batch[node];
    float4 v = h4[(size_t)node * 16 + f4];
    float* pp = psum + (size_t)b * FH + f4 * 4;
    atomicAdd(pp + 0, v.x);
    atomicAdd(pp + 1, v.y);
    atomicAdd(pp + 2, v.z);
    atomicAdd(pp + 3, v.w);
    if (f4 == 0) atomicAdd(&pcnt[b], 1.0f);
}

__global__ void gcn_head(const float* __restrict__ psum, const float* __restrict__ pcnt,
                         const float* __restrict__ Wl, const float* __restrict__ bl,
                         float* __restrict__ out) {
    int t = blockIdx.x * blockDim.x + threadIdx.x;
    if (t >= BB * CC) return;
    int b = t >> 1, c = t & 1;
    float inv = 1.0f / fmaxf(pcnt[b], 1.0f);
    float acc = bl[c];
    for (int f = 0; f < FH; ++f)
        acc += psum[(size_t)b * FH + f] * inv * Wl[f * CC + c];
    out[t] = acc;
}

// ---------------- launch ----------------

extern "C" void kernel_launch(void* const* d_in, const int* in_sizes, int n_in,
                              void* d_out, int out_size, void* d_ws, size_t ws_size,
                              hipStream_t stream) {
    const float* x    = (const float*)d_in[0];
    const int*   eidx = (const int*)d_in[1];     // [2, E] flat
    const int*   batch= (const int*)d_in[2];
    const float* W1 = (const float*)d_in[3];  const float* b1 = (const float*)d_in[4];
    const float* W2 = (const float*)d_in[5];  const float* b2 = (const float*)d_in[6];
    const float* W3 = (const float*)d_in[7];  const float* b3 = (const float*)d_in[8];
    const float* Wl = (const float*)d_in[9];  const float* bl = (const float*)d_in[10];
    float* out = (float*)d_out;

    const int* src = eidx;
    const int* dst = eidx + EE;

    // workspace carve-out
    char* ws = (char*)d_ws;
    size_t off = 0;
    auto carve = [&](size_t bytes) { char* p = ws + off; off = (off + bytes + 255) & ~(size_t)255; return p; };
    float* dinv = (float*)carve((size_t)NN * 4);
    float* norm = (float*)carve((size_t)EE * 4);
    float* hbuf = (float*)carve((size_t)NN * FH * 4);
    float* agg  = (float*)carve((size_t)NN * FH * 4);
    float* psum = (float*)carve((size_t)BB * FH * 4);
    float* pcnt = (float*)carve((size_t)BB * 4);
    (void)ws_size;

    const int T = 256;
    auto cdiv = [](long long a, long long b) { return (int)((a + b - 1) / b); };

    // degree -> dinv_sqrt -> per-edge norm (reused for all 3 layers)
    gcn_zero_f32<<<cdiv(NN, T), T, 0, stream>>>(dinv, NN);
    gcn_deg_accum<<<cdiv(EE, T), T, 0, stream>>>(dst, dinv, EE);
    gcn_deg_finalize<<<cdiv(NN, T), T, 0, stream>>>(dinv, NN);
    gcn_edge_norm<<<cdiv(EE, T), T, 0, stream>>>(src, dst, dinv, norm, EE);

    const int gemm_grid  = cdiv((long long)(NN / 16) * 32, T);   // one wave per 16-row tile
    const int node4_grid = cdiv((long long)NN * 16, T);          // 16 float4-threads per node
    const int edge4_grid = cdiv((long long)EE * 16, T);          // 16 float4-threads per edge
    const int relu_grid  = cdiv((long long)NN * FH / 4, T);

    const float4* hbuf4 = (const float4*)hbuf;
    float4* agg4 = (float4*)agg;

    // layer 1: x -> agg
    gcn_gemm64_wmma<<<gemm_grid, T, 0, stream>>>(x, W1, hbuf, NN);
    gcn_agg_init<<<node4_grid, T, 0, stream>>>(hbuf4, dinv, (const float4*)b1, agg4, NN);
    gcn_scatter<<<edge4_grid, T, 0, stream>>>(hbuf4, src, dst, norm, agg, EE);
    gcn_relu4<<<relu_grid, T, 0, stream>>>(agg4, NN * FH / 4);

    // layer 2: agg -> agg
    gcn_gemm64_wmma<<<gemm_grid, T, 0, stream>>>(agg, W2, hbuf, NN);
    gcn_agg_init<<<node4_grid, T, 0, stream>>>(hbuf4, dinv, (const float4*)b2, agg4, NN);
    gcn_scatter<<<edge4_grid, T, 0, stream>>>(hbuf4, src, dst, norm, agg, EE);
    gcn_relu4<<<relu_grid, T, 0, stream>>>(agg4, NN * FH / 4);

    // layer 3 (no relu)
    gcn_gemm64_wmma<<<gemm_grid, T, 0, stream>>>(agg, W3, hbuf, NN);
    gcn_agg_init<<<node4_grid, T, 0, stream>>>(hbuf4, dinv, (const float4*)b3, agg4, NN);
    gcn_scatter<<<edge4_grid, T, 0, stream>>>(hbuf4, src, dst, norm, agg, EE);

    // pooling + head
    gcn_zero_f32<<<cdiv(BB * FH, T), T, 0, stream>>>(psum, BB * FH);
    gcn_zero_f32<<<cdiv(BB, T), T, 0, stream>>>(pcnt, BB);
    gcn_pool_accum<<<node4_grid, T, 0, stream>>>((const float4*)agg, batch, psum, pcnt, NN);
    gcn_head<<<cdiv(BB * CC, T), T, 0, stream>>>(psum, pcnt, Wl, bl, out);
}